// EntropyMetircs_1d_83288005804274
// MI455X (gfx1250) — compile-verified
//
#include <hip/hip_runtime.h>
#include <hip/hip_bf16.h>

typedef float f32x4 __attribute__((ext_vector_type(4)));
typedef float v2f   __attribute__((ext_vector_type(2)));
typedef float v8f   __attribute__((ext_vector_type(8)));

#define TPB   256
#define WAVES (TPB / 32)
#define NBINS 256
#define BPS   8   // blocks per slice

__global__ void zero_kernel(unsigned* __restrict__ p, int n) {
    int i = blockIdx.x * blockDim.x + threadIdx.x;
    if (i < n) p[i] = 0u;
}

// One pass over 1/BPS of a slice. LDS-privatized per-wave histograms, NT 128-bit loads.
// If `gate` != nullptr: early-exit (scalar branch) when the recorded global max < 1.0.
// If `do_max`: track block max and atomicMax it into maxbits (bit-max == float-max for x>=0).
__global__ void hist_kernel(const float* __restrict__ x,
                            unsigned* __restrict__ counts,
                            unsigned* __restrict__ maxbits,
                            const unsigned* __restrict__ gate,
                            float scale, int hw, int bps, int do_max) {
    if (gate) {
        float m = __uint_as_float(*gate);   // scalar (uniform) load + branch
        if (m < 1.0f) return;
    }

    __shared__ unsigned hist[WAVES * NBINS];
    __shared__ float    wmax[WAVES];

    const int tid  = threadIdx.x;
    const int wave = tid >> 5;
    const int lane = tid & 31;

    for (int i = tid; i < WAVES * NBINS; i += TPB) hist[i] = 0u;
    __syncthreads();

    const int slice = blockIdx.x / bps;
    const int sub   = blockIdx.x % bps;
    const int f4n   = hw / (4 * bps);   // float4 elements handled by this block

    const f32x4* __restrict__ p =
        (const f32x4*)(x + (size_t)slice * (size_t)hw) + (size_t)sub * (size_t)f4n;

    unsigned* __restrict__ h = hist + wave * NBINS;
    float lmax = 0.0f;

    for (int i = tid; i < f4n; i += TPB) {
        f32x4 v = __builtin_nontemporal_load(p + i);   // global_load_b128, TH=NT
        if (do_max) {
            lmax = fmaxf(lmax, fmaxf(fmaxf(v.x, v.y), fmaxf(v.z, v.w)));
        }
        int b0 = min(255, max(0, (int)(v.x * scale)));
        int b1 = min(255, max(0, (int)(v.y * scale)));
        int b2 = min(255, max(0, (int)(v.z * scale)));
        int b3 = min(255, max(0, (int)(v.w * scale)));
        atomicAdd(&h[b0], 1u);   // ds_add_u32
        atomicAdd(&h[b1], 1u);
        atomicAdd(&h[b2], 1u);
        atomicAdd(&h[b3], 1u);
    }

    if (do_max) {
        for (int off = 16; off > 0; off >>= 1)
            lmax = fmaxf(lmax, __shfl_xor(lmax, off, 32));   // wave32 reduce
        if (lane == 0) wmax[wave] = lmax;
    }

    __syncthreads();

    // merge wave-private histograms -> one global atomic per bin per block
    for (int b = tid; b < NBINS; b += TPB) {
        unsigned s = 0;
        #pragma unroll
        for (int w = 0; w < WAVES; ++w) s += hist[w * NBINS + b];
        if (s) atomicAdd(&counts[(size_t)slice * NBINS + b], s);
    }

    if (do_max && tid == 0) {
        float m = wmax[0];
        #pragma unroll
        for (int w = 1; w < WAVES; ++w) m = fmaxf(m, wmax[w]);
        atomicMax(maxbits, __float_as_uint(m));
    }
}

// One block, 256 threads. Thread t computes entropy of slice t, then wave 0 sums the
// 256 entropies with v_wmma_f32_16x16x4_f32 (A = ones => layout-agnostic column sums).
__global__ void entropy_kernel(const unsigned* __restrict__ countsA,
                               const unsigned* __restrict__ countsB,
                               const unsigned* __restrict__ maxbits,
                               float* __restrict__ out,
                               int nslices, float inv_hw) {
    __shared__ float ent[256];
    const int t = threadIdx.x;

    const unsigned* counts =
        (__uint_as_float(*maxbits) < 1.0f) ? countsA : countsB;

    float e = 0.0f;
    if (t < nslices) {
        const unsigned* c = counts + (size_t)t * NBINS;
        for (int b = 0; b < NBINS; ++b) {
            unsigned cb = c[b];
            if (cb) {
                float pp = (float)cb * inv_hw;
                e -= pp * log2f(pp);
            }
        }
    }
    ent[t] = e;
    __syncthreads();

    if (t < 32) {
        // A = 16x4 ones (2 VGPRs), B = 4x16 data chunk (64 values in 2 VGPRs).
        // D = A*B + C  =>  every D element is a column-sum of B; sum of all 256
        // D elements = 16 * sum(B) regardless of lane layout.
        v2f a; a.x = 1.0f; a.y = 1.0f;
        v8f c = {};
        #pragma unroll
        for (int chunk = 0; chunk < 4; ++chunk) {
            v2f b;
            b.x = ent[chunk * 64 + t];
            b.y = ent[chunk * 64 + 32 + t];
            c = __builtin_amdgcn_wmma_f32_16x16x4_f32(
                    false, a, false, b, (short)0, c, false, false);
        }
        float s = c[0] + c[1] + c[2] + c[3] + c[4] + c[5] + c[6] + c[7];
        for (int off = 16; off > 0; off >>= 1)
            s += __shfl_xor(s, off, 32);
        if (t == 0) out[0] = s / (16.0f * (float)nslices);
    }
}

extern "C" void kernel_launch(void* const* d_in, const int* in_sizes, int n_in,
                              void* d_out, int out_size, void* d_ws, size_t ws_size,
                              hipStream_t stream) {
    const float* x = (const float*)d_in[0];
    float* out = (float*)d_out;

    const int HW = 1024 * 1024;
    const int nslices = in_sizes[0] / HW;   // 256

    unsigned* ws      = (unsigned*)d_ws;
    unsigned* maxbits = ws;                       // 1 word (+63 pad)
    unsigned* countsA = ws + 64;                  // nslices*256 words
    unsigned* countsB = countsA + (size_t)nslices * NBINS;

    const int zeroN = 64 + 2 * nslices * NBINS;
    zero_kernel<<<(zeroN + TPB - 1) / TPB, TPB, 0, stream>>>(ws, zeroN);

    dim3 grid(nslices * BPS);
    // pass 1: x*255 histogram + fused global max (the common path)
    hist_kernel<<<grid, TPB, 0, stream>>>(x, countsA, maxbits, nullptr,
                                          255.0f, HW, BPS, 1);
    // pass 2: raw-scale histogram; scalar early-exit when max < 1.0
    hist_kernel<<<grid, TPB, 0, stream>>>(x, countsB, nullptr, maxbits,
                                          1.0f, HW, BPS, 0);

    entropy_kernel<<<1, 256, 0, stream>>>(countsA, countsB, maxbits, out,
                                          nslices, 1.0f / (float)HW);
}